// SpOptNetEq_37082747633749
// MI455X (gfx1250) — compile-verified
//
#include <hip/hip_runtime.h>

// Problem constants (sudoku OptNet QP, n=3): fixed by the reference.
#define NX       729      // number of variables
#define NEQ      250      // number of equality constraints
#define NXP      736      // padded vector length (Hinv pad MUST stay zero)
#define NEQP     256      // padded neq
#define LDK      257      // row stride of K in LDS (odd -> bank-conflict friendly)
#define NITER    15
#define NTHREADS 256      // 8 wave32s
#define KSTEPS   ((NX - 1) / 4)   // 182 fully in-bounds k-steps (n = 0..727)

typedef float v2f __attribute__((ext_vector_type(2)));
typedef float v8f __attribute__((ext_vector_type(8)));

__device__ __forceinline__ float warpSum(float v) {
#pragma unroll
  for (int o = 16; o > 0; o >>= 1) v += __shfl_down(v, o, 32);
  return v;
}
__device__ __forceinline__ float warpMin(float v) {
#pragma unroll
  for (int o = 16; o > 0; o >>= 1) v = fminf(v, __shfl_down(v, o, 32));
  return v;
}
__device__ float blockSum(float v, float* scr) {
  const int lane = threadIdx.x & 31, wid = threadIdx.x >> 5;
  v = warpSum(v);
  __syncthreads();
  if (lane == 0) scr[wid] = v;
  __syncthreads();
  if (threadIdx.x == 0) {
    float a = 0.f;
#pragma unroll
    for (int i = 0; i < NTHREADS / 32; ++i) a += scr[i];
    scr[0] = a;
  }
  __syncthreads();
  return scr[0];
}
__device__ float blockMin(float v, float* scr) {
  const int lane = threadIdx.x & 31, wid = threadIdx.x >> 5;
  v = warpMin(v);
  __syncthreads();
  if (lane == 0) scr[wid] = v;
  __syncthreads();
  if (threadIdx.x == 0) {
    float a = scr[0];
#pragma unroll
    for (int i = 1; i < NTHREADS / 32; ++i) a = fminf(a, scr[i]);
    scr[0] = a;
  }
  __syncthreads();
  return scr[0];
}

// In-LDS Cholesky of K[0:NEQ,0:NEQ] (lower triangle, in place).
__device__ void cholesky_lds(float* Ks) {
  const int tid = threadIdx.x;
  for (int j = 0; j < NEQ; ++j) {
    const float d   = Ks[j * LDK + j];   // valid: trailing update of j-1 was barriered
    const float ljj = sqrtf(d);
    __syncthreads();                     // everyone has read d before tid0 overwrites
    if (tid == 0) Ks[j * LDK + j] = ljj;
    const float linv = 1.0f / ljj;
    for (int i = j + 1 + tid; i < NEQ; i += NTHREADS) Ks[i * LDK + j] *= linv;
    __syncthreads();                     // scaled column visible
    for (int i = j + 1 + tid; i < NEQ; i += NTHREADS) {
      const float lij = Ks[i * LDK + j];
      for (int c = j + 1; c <= i; ++c) Ks[i * LDK + c] -= lij * Ks[c * LDK + j];
    }
    __syncthreads();                     // trailing update done
  }
}

// Solve  diag(H) dz + A^T dnu = r1 ;  A dz = -rnu   via Schur complement:
//   dnu = K^{-1} (A (Hinv.*r1) + rnu),  dz = Hinv .* (r1 - A^T dnu)
// K's Cholesky factor (lower) already sits in Ks.
__device__ void kkt_solve(const float* __restrict__ Ag, const float* Ks,
                          const float* Hinv, const float* r1,
                          const float* rnu, float* dnu, float* dz) {
  const int tid = threadIdx.x, lane = tid & 31, wave = tid >> 5;
  // rhs = A (Hinv.*r1) + rnu   (wave-per-row, coalesced reads of L2-resident A)
  for (int e = wave; e < NEQ; e += NTHREADS / 32) {
    const float* Ar = Ag + e * NX;
    float acc = 0.f;
    for (int n = lane; n < NX; n += 32) acc += Ar[n] * Hinv[n] * r1[n];
    acc = warpSum(acc);
    if (lane == 0) dnu[e] = acc + rnu[e];
  }
  __syncthreads();
  // Triangular solves by a single wave: wave32 in-order LDS => no block barriers
  if (wave == 0) {
    for (int j = 0; j < NEQ; ++j) {              // L y = rhs
      const float xj = dnu[j] / Ks[j * LDK + j];
      if (lane == 0) dnu[j] = xj;
      for (int i = j + 1 + lane; i < NEQ; i += 32) dnu[i] -= Ks[i * LDK + j] * xj;
    }
    for (int j = NEQ - 1; j >= 0; --j) {         // L^T x = y
      const float xj = dnu[j] / Ks[j * LDK + j];
      if (lane == 0) dnu[j] = xj;
      for (int i = lane; i < j; i += 32) dnu[i] -= Ks[j * LDK + i] * xj;
    }
  }
  __syncthreads();
  // dz = Hinv .* (r1 - A^T dnu)
  for (int n = tid; n < NX; n += NTHREADS) {
    float acc = 0.f;
    for (int e = 0; e < NEQ; ++e) acc += Ag[e * NX + n] * dnu[e];
    dz[n] = Hinv[n] * (r1[n] - acc);
  }
  __syncthreads();
}

__global__ __launch_bounds__(NTHREADS) void qp_ipm_kernel(
    const float* __restrict__ puzzles, const float* __restrict__ Ag,
    const float* __restrict__ qvg, const float* __restrict__ bg,
    float* __restrict__ out) {
  extern __shared__ float smem[];
  float* Ks   = smem;                 // 256 x LDK  (Schur matrix / Cholesky factor)
  float* vecs = Ks + 256 * LDK;
  float* p    = vecs + 0 * NXP;
  float* z    = vecs + 1 * NXP;
  float* s    = vecs + 2 * NXP;
  float* lam  = vecs + 3 * NXP;
  float* qv   = vecs + 4 * NXP;
  float* Hinv = vecs + 5 * NXP;       // pad [NX..NXP) stays 0 forever (k-tail trick)
  float* rz   = vecs + 6 * NXP;
  float* rlam = vecs + 7 * NXP;
  float* vX   = vecs + 8 * NXP;       // r1_aff -> rs
  float* vY   = vecs + 9 * NXP;       // dz_aff -> ds_aff -> r1_cor
  float* vZ   = vecs + 10 * NXP;      // dlam_aff -> dz_cor
  float* nuv  = vecs + 11 * NXP;      // NEQP
  float* rnu  = nuv + NEQP;
  float* dnu  = rnu + NEQP;
  float* scr  = dnu + NEQP;           // 32 floats reduction scratch

  const int tid  = threadIdx.x;
  const int lane = tid & 31;
  const int wave = tid >> 5;
  const int bidx = blockIdx.x;

  // ---- init state: z=0, s=1, lam=1, nu=0, p = -puzzles[b] ----
  for (int i = tid; i < NX; i += NTHREADS) {
    p[i]   = -puzzles[(size_t)bidx * NX + i];
    qv[i]  = qvg[i];
    z[i]   = 0.f;
    s[i]   = 1.f;
    lam[i] = 1.f;
  }
  for (int i = NX + tid; i < NXP; i += NTHREADS) Hinv[i] = 0.f;  // permanent zero pad
  for (int e = tid; e < NEQP; e += NTHREADS) nuv[e] = 0.f;
  __syncthreads();

  for (int it = 0; it < NITER; ++it) {
    // ---- diagonal and residuals ----
    for (int i = tid; i < NX; i += NTHREADS) {
      const float si = s[i], li = lam[i];
      Hinv[i] = 1.f / (qv[i] + li / si);
      rlam[i] = si - z[i];
    }
    __syncthreads();
    // rz = qv*z + p - lam + A^T nu
    for (int n = tid; n < NX; n += NTHREADS) {
      float acc = 0.f;
      for (int e = 0; e < NEQ; ++e) acc += Ag[e * NX + n] * nuv[e];
      rz[n] = qv[n] * z[n] + p[n] - lam[n] + acc;
    }
    // rnu = A z - b
    for (int e = wave; e < NEQ; e += NTHREADS / 32) {
      const float* Ar = Ag + e * NX;
      float acc = 0.f;
      for (int n = lane; n < NX; n += 32) acc += Ar[n] * z[n];
      acc = warpSum(acc);
      if (lane == 0) rnu[e] = acc - bg[e];
    }
    // mu = mean(s*lam)
    float part = 0.f;
    for (int i = tid; i < NX; i += NTHREADS) part += s[i] * lam[i];
    const float mu = blockSum(part, scr) * (1.f / NX);
    __syncthreads();

    // ---- K = A diag(Hinv) A^T via V_WMMA_F32_16X16X4_F32, K resident in LDS ----
    // 2x2 tile blocking over the upper triangle of the 8x8 super-tile grid.
    // No guards in the inner loop: A row pointers are clamped (garbage lands in
    // K's pad rows/cols >=250, never read), and pad-k columns contribute exactly
    // zero because the A-operand carries the Hinv scale and Hinv[729..735]==0.
    // The k-loop induction is UNIFORM (kk) so the compiler keeps a scalar branch
    // and EXEC stays all-ones around every WMMA by construction.
    {
      const int m  = lane & 15;          // row-in-tile (A) / col-in-tile (B)
      const int kh = (lane >> 4) << 1;   // lanes 0-15 -> K=0,1 ; lanes 16-31 -> K=2,3
      const int ro = (lane >> 4) << 3;   // C/D row offset: +8 for upper half-wave
      for (int st = wave; st < 36; st += NTHREADS / 32) {
        int et2 = 0, tt = st;
        while (tt >= (8 - et2)) { tt -= (8 - et2); ++et2; }
        const int ft2 = et2 + tt;
        const int e0 = et2 << 5, f0 = ft2 << 5;      // 32-wide super-tile bases
        const int re0 = e0 + m,      re1 = e0 + 16 + m;
        const int rf0 = f0 + m,      rf1 = f0 + 16 + m;
        const float* Ae0 = Ag + (re0 < NEQ ? re0 : NEQ - 1) * NX;
        const float* Ae1 = Ag + (re1 < NEQ ? re1 : NEQ - 1) * NX;
        const float* Af0 = Ag + (rf0 < NEQ ? rf0 : NEQ - 1) * NX;
        const float* Af1 = Ag + (rf1 < NEQ ? rf1 : NEQ - 1) * NX;
        v8f c00 = {}, c01 = {}, c10 = {}, c11 = {};
        // main loop: n = 0..727, fully in bounds, uniform trip count
        for (int kk = 0; kk < KSTEPS; ++kk) {
          const int ka = (kk << 2) + kh;
          const float h0 = Hinv[ka], h1 = Hinv[ka + 1];
          v2f a0, a1, b0, b1;
          a0.x = Ae0[ka] * h0;  a0.y = Ae0[ka + 1] * h1;
          a1.x = Ae1[ka] * h0;  a1.y = Ae1[ka + 1] * h1;
          b0.x = Af0[ka];       b0.y = Af0[ka + 1];
          b1.x = Af1[ka];       b1.y = Af1[ka + 1];
          c00 = __builtin_amdgcn_wmma_f32_16x16x4_f32(false, a0, false, b0, (short)0, c00, false, false);
          c01 = __builtin_amdgcn_wmma_f32_16x16x4_f32(false, a0, false, b1, (short)0, c01, false, false);
          c10 = __builtin_amdgcn_wmma_f32_16x16x4_f32(false, a1, false, b0, (short)0, c10, false, false);
          c11 = __builtin_amdgcn_wmma_f32_16x16x4_f32(false, a1, false, b1, (short)0, c11, false, false);
        }
        // tail step: k0 = 728; clamp indices, Hinv zero-pad kills k>=729 terms
        {
          const int ka = 728 + kh;                       // 728 or 730
          const int kc0 = (ka < NX) ? ka : NX - 1;
          const float h0 = Hinv[ka], h1 = Hinv[ka + 1];  // <= 731 < NXP; pad = 0
          v2f a0, a1, b0, b1;
          a0.x = Ae0[kc0] * h0;  a0.y = Ae0[NX - 1] * h1;
          a1.x = Ae1[kc0] * h0;  a1.y = Ae1[NX - 1] * h1;
          b0.x = Af0[kc0];       b0.y = Af0[NX - 1];
          b1.x = Af1[kc0];       b1.y = Af1[NX - 1];
          c00 = __builtin_amdgcn_wmma_f32_16x16x4_f32(false, a0, false, b0, (short)0, c00, false, false);
          c01 = __builtin_amdgcn_wmma_f32_16x16x4_f32(false, a0, false, b1, (short)0, c01, false, false);
          c10 = __builtin_amdgcn_wmma_f32_16x16x4_f32(false, a1, false, b0, (short)0, c10, false, false);
          c11 = __builtin_amdgcn_wmma_f32_16x16x4_f32(false, a1, false, b1, (short)0, c11, false, false);
        }
        // store tiles + symmetric mirrors (duplicates are bitwise identical)
#pragma unroll
        for (int r = 0; r < 8; ++r) {
          const int r0 = e0 + r + ro, r1 = r0 + 16;
          const int q0 = f0 + m,      q1 = q0 + 16;
          const float v00 = c00[r], v01 = c01[r], v10 = c10[r], v11 = c11[r];
          Ks[r0 * LDK + q0] = v00;  Ks[q0 * LDK + r0] = v00;
          Ks[r0 * LDK + q1] = v01;  Ks[q1 * LDK + r0] = v01;
          Ks[r1 * LDK + q0] = v10;  Ks[q0 * LDK + r1] = v10;
          Ks[r1 * LDK + q1] = v11;  Ks[q1 * LDK + r1] = v11;
        }
      }
    }
    __syncthreads();
    for (int i = tid; i < NEQ; i += NTHREADS) Ks[i * LDK + i] += 1e-6f;
    __syncthreads();

    cholesky_lds(Ks);

    // ---- affine (predictor) direction: r1 = -rz - lam + (lam/s)*rlam ----
    for (int i = tid; i < NX; i += NTHREADS) {
      const float dd = lam[i] / s[i];
      vX[i] = -rz[i] - lam[i] + dd * rlam[i];
    }
    __syncthreads();
    kkt_solve(Ag, Ks, Hinv, vX, rnu, dnu, vY);   // vY = dz_aff

    float stp = __builtin_inff();
    for (int i = tid; i < NX; i += NTHREADS) {
      const float dsa = vY[i] - rlam[i];
      const float dla = -(lam[i] * (s[i] + dsa)) / s[i];
      vY[i] = dsa;                                // ds_aff
      vZ[i] = dla;                                // dlam_aff
      if (dsa < 0.f) stp = fminf(stp, -s[i] / dsa);
      if (dla < 0.f) stp = fminf(stp, -lam[i] / dla);
    }
    const float a_aff = fminf(1.f, blockMin(stp, scr));
    float part2 = 0.f;
    for (int i = tid; i < NX; i += NTHREADS)
      part2 += (s[i] + a_aff * vY[i]) * (lam[i] + a_aff * vZ[i]);
    const float mu_aff = blockSum(part2, scr) * (1.f / NX);
    float sg = mu_aff / mu;
    sg = sg * sg * sg;
    const float sigma = fminf(1.f, fmaxf(0.f, sg));

    // ---- corrector + centering: rs = lam*s + ds_a*dlam_a - sigma*mu ----
    for (int i = tid; i < NX; i += NTHREADS) {
      const float rs_i = lam[i] * s[i] + vY[i] * vZ[i] - sigma * mu;
      vX[i] = rs_i;
      vY[i] = -rz[i] - rs_i / s[i] + (lam[i] / s[i]) * rlam[i];  // r1_cor
    }
    __syncthreads();
    kkt_solve(Ag, Ks, Hinv, vY, rnu, dnu, vZ);   // vZ = dz

    float stp2 = __builtin_inff();
    for (int i = tid; i < NX; i += NTHREADS) {
      const float dsi = vZ[i] - rlam[i];
      const float dli = -(vX[i] + lam[i] * dsi) / s[i];
      if (dsi < 0.f) stp2 = fminf(stp2, -s[i] / dsi);
      if (dli < 0.f) stp2 = fminf(stp2, -lam[i] / dli);
    }
    const float alpha = fminf(1.f, 0.995f * blockMin(stp2, scr));
    for (int i = tid; i < NX; i += NTHREADS) {
      const float dsi = vZ[i] - rlam[i];
      const float dli = -(vX[i] + lam[i] * dsi) / s[i];
      z[i]   += alpha * vZ[i];
      s[i]   += alpha * dsi;
      lam[i] += alpha * dli;
    }
    for (int e = tid; e < NEQ; e += NTHREADS) nuv[e] += alpha * dnu[e];
    __syncthreads();
  }

  for (int i = tid; i < NX; i += NTHREADS)
    out[(size_t)bidx * NX + i] = z[i];
}

extern "C" void kernel_launch(void* const* d_in, const int* in_sizes, int n_in,
                              void* d_out, int out_size, void* d_ws, size_t ws_size,
                              hipStream_t stream) {
  (void)n_in; (void)d_ws; (void)ws_size; (void)out_size;
  const float* puzzles = (const float*)d_in[0];
  const float* A       = (const float*)d_in[1];
  const float* qv      = (const float*)d_in[2];
  const float* b       = (const float*)d_in[3];
  float* out = (float*)d_out;

  const int B = in_sizes[0] / NX;   // 256 batch elements -> 256 workgroups

  const size_t shmem =
      ((size_t)256 * LDK + 11 * NXP + 3 * NEQP + 32) * sizeof(float);  // ~292 KB
  (void)hipFuncSetAttribute((const void*)qp_ipm_kernel,
                            hipFuncAttributeMaxDynamicSharedMemorySize,
                            (int)shmem);
  qp_ipm_kernel<<<B, NTHREADS, shmem, stream>>>(puzzles, A, qv, b, out);
}